// NNkNNRegression_80659485819606
// MI455X (gfx1250) — compile-verified
//
#include <hip/hip_runtime.h>
#include <stdint.h>
#include <math.h>

#define B_  128
#define N_  65536
#define D_  64
#define K_  32
#define CHUNKS 32
#define ROWS_PER_CHUNK (N_/CHUNKS)           // 2048
#define TILE_ROWS 64
#define TILES_PER_CHUNK (ROWS_PER_CHUNK/TILE_ROWS) // 32
#define WAVES 8
#define ROWS_PER_WAVE_TILE (TILE_ROWS/WAVES) // 8
#define NCAND (CHUNKS*WAVES*K_)              // 8192 candidates per query

typedef float v2f __attribute__((ext_vector_type(2)));
typedef float v8f __attribute__((ext_vector_type(8)));

__device__ __forceinline__ float softplus_f(float x) {
  return (x > 20.f) ? x : log1pf(__expf(x));
}

// ---- CDNA5 async global->LDS copy (ASYNCcnt-tracked) ----
__device__ __forceinline__ void async_copy_b128(uint32_t lds_off, const void* gaddr) {
  asm volatile("global_load_async_to_lds_b128 %0, %1, off"
               :: "v"(lds_off), "v"(gaddr) : "memory");
}
__device__ __forceinline__ void wait_asynccnt0() {
  asm volatile("s_wait_asynccnt 0" ::: "memory");
}

// =====================================================================
// Phase 1: weighted-L1 distance, per-(lane,wave,chunk) top-32 on vd = -dist
// grid = (CHUNKS, 4 query-groups), block = 256 (8 waves, lane = query)
// =====================================================================
__global__ __launch_bounds__(256) void k_dist_topk(
    const float* __restrict__ q, const float* __restrict__ X,
    const float* __restrict__ fw,
    float* __restrict__ candVal, int* __restrict__ candIdx)
{
  __shared__ float tile[2][TILE_ROWS * D_];  // 2 x 16KB double buffer
  __shared__ float wsh[D_];                  // softplus(fw), computed once

  const int tid  = threadIdx.x;
  const int lane = tid & 31;
  const int wave = tid >> 5;
  const int grp  = blockIdx.y;   // 0..3
  const int chunk= blockIdx.x;   // 0..31
  const int b    = grp * 32 + lane;
  const int rowBase = chunk * ROWS_PER_CHUNK;

  // Kick off first tile immediately (16KB contiguous; 256 thr x 4 x b128)
  {
    const char* src = (const char*)(X + (size_t)rowBase * D_);
    uint32_t l0 = (uint32_t)(uintptr_t)(&tile[0][0]);
    #pragma unroll
    for (int j = 0; j < 4; ++j)
      async_copy_b128(l0 + (uint32_t)tid * 16u + (uint32_t)j * 4096u,
                      src + tid * 16 + j * 4096);
  }

  // One softplus expansion per thread, shared through LDS
  if (tid < D_) wsh[tid] = softplus_f(fw[tid]);
  __syncthreads();

  // Pre-scaled query in registers: qs[d] = w[d]*q[b][d]  (w|q-x| == |wq-wx|, w>0)
  float qs[D_];
  {
    const float4* qp = (const float4*)(q + (size_t)b * D_);
    #pragma unroll
    for (int k4 = 0; k4 < 16; ++k4) {
      float4 q4 = qp[k4];
      qs[4*k4+0] = wsh[4*k4+0] * q4.x;
      qs[4*k4+1] = wsh[4*k4+1] * q4.y;
      qs[4*k4+2] = wsh[4*k4+2] * q4.z;
      qs[4*k4+3] = wsh[4*k4+3] * q4.w;
    }
  }
  // Per-thread weight float4 for tile scaling (element phase = tid%16)
  float4 w4c;
  {
    int d0 = (tid & 15) * 4;
    w4c.x = wsh[d0+0]; w4c.y = wsh[d0+1]; w4c.z = wsh[d0+2]; w4c.w = wsh[d0+3];
  }

  // Sorted-ascending per-lane top-K on vd = -dist (tv[0] = threshold)
  float tv[K_]; int ti[K_];
  #pragma unroll
  for (int k = 0; k < K_; ++k) { tv[k] = -3.0e38f; ti[k] = 0; }

  for (int t = 0; t < TILES_PER_CHUNK; ++t) {
    const int cur = t & 1;
    wait_asynccnt0();
    __syncthreads();

    // Scale current tile in LDS by w (once per tile instead of per row)
    {
      float4* p = (float4*)&tile[cur][0];
      #pragma unroll
      for (int j = 0; j < 4; ++j) {
        float4 v = p[tid + 256 * j];
        v.x *= w4c.x; v.y *= w4c.y; v.z *= w4c.z; v.w *= w4c.w;
        p[tid + 256 * j] = v;
      }
    }
    __syncthreads();

    // Prefetch next tile into the other buffer
    if (t + 1 < TILES_PER_CHUNK) {
      const char* src = (const char*)(X + (size_t)(rowBase + (t + 1) * TILE_ROWS) * D_);
      uint32_t l0 = (uint32_t)(uintptr_t)(&tile[cur ^ 1][0]);
      #pragma unroll
      for (int j = 0; j < 4; ++j)
        async_copy_b128(l0 + (uint32_t)tid * 16u + (uint32_t)j * 4096u,
                        src + tid * 16 + j * 4096);
    }

    // Each wave consumes 8 rows; X reads are wave-uniform LDS broadcasts
    #pragma unroll
    for (int r = 0; r < ROWS_PER_WAVE_TILE; ++r) {
      const int rloc = wave * ROWS_PER_WAVE_TILE + r;
      const float4* xr = (const float4*)&tile[cur][rloc * D_];
      float dist = 0.f;
      #pragma unroll
      for (int k4 = 0; k4 < 16; ++k4) {
        float4 x4 = xr[k4];
        dist += fabsf(qs[4*k4+0] - x4.x);
        dist += fabsf(qs[4*k4+1] - x4.y);
        dist += fabsf(qs[4*k4+2] - x4.z);
        dist += fabsf(qs[4*k4+3] - x4.w);
      }
      const float vd = -dist;                 // exp is monotone: select on -dist
      const int gRow = rowBase + t * TILE_ROWS + rloc;
      if (vd > tv[0]) {
        tv[0] = vd; ti[0] = gRow;
        #pragma unroll
        for (int j = 0; j < K_ - 1; ++j) {    // one bubble pass re-sorts
          bool sw = tv[j] > tv[j+1];
          float fv = tv[j]; int fi = ti[j];
          tv[j]   = sw ? tv[j+1] : tv[j];
          ti[j]   = sw ? ti[j+1] : ti[j];
          tv[j+1] = sw ? fv : tv[j+1];
          ti[j+1] = sw ? fi : ti[j+1];
        }
      }
    }
  }

  // Emit partial top-K: cand[b][chunk][wave][k]  (values are -dist)
  size_t base = (((size_t)b * CHUNKS + chunk) * WAVES + wave) * K_;
  #pragma unroll
  for (int k = 0; k < K_; ++k) { candVal[base + k] = tv[k]; candIdx[base + k] = ti[k]; }
}

// =====================================================================
// Phase 2: per-query merge of 8192 candidates -> top-32, act=exp(b*vd),
// softmax, y-gather.  grid = 128 (one block per query), block = 256
// =====================================================================
__global__ __launch_bounds__(256) void k_select(
    const float* __restrict__ candVal, const int* __restrict__ candIdx,
    const float* __restrict__ y, const float* __restrict__ beta,
    float* __restrict__ pred)
{
  __shared__ float sv[NCAND];
  __shared__ float rv[256];
  __shared__ int   rp[256];
  __shared__ float topv[K_];
  __shared__ int   topi[K_];

  const int b = blockIdx.x, tid = threadIdx.x;
  const float* cv = candVal + (size_t)b * NCAND;
  for (int i = tid; i < NCAND; i += 256) sv[i] = cv[i];
  __syncthreads();

  for (int k = 0; k < K_; ++k) {
    float m = -INFINITY; int p = 0;
    for (int i = tid; i < NCAND; i += 256) { float v = sv[i]; if (v > m) { m = v; p = i; } }
    rv[tid] = m; rp[tid] = p;
    __syncthreads();
    for (int s = 128; s > 0; s >>= 1) {
      if (tid < s && rv[tid + s] > rv[tid]) { rv[tid] = rv[tid + s]; rp[tid] = rp[tid + s]; }
      __syncthreads();
    }
    if (tid == 0) {
      int p0 = rp[0];
      topv[k] = rv[0];
      topi[k] = candIdx[(size_t)b * NCAND + p0];
      sv[p0] = -INFINITY;
    }
    __syncthreads();
  }

  if (tid == 0) {
    const float bsp = softplus_f(beta[0]);
    const float amax = __expf(bsp * topv[0]);   // largest activation
    float s = 0.f, acc = 0.f;
    for (int k = 0; k < K_; ++k) {
      float a = __expf(bsp * topv[k]);          // act = exp(-b*dist)
      float e = __expf(a - amax);               // softmax over activations
      s += e;
      acc += e * y[topi[k]];                    // OUT == 1
    }
    pred[b] = acc / s;
  }
}

// =====================================================================
// Phase 3: MLP head via v_wmma_f32_16x16x4_f32.
// h = relu(pred @ W1.T + b1) [128,32]; out = h @ W2.T + b2 [128,1]
// One block, 8 waves; wave w owns batch tile [16w, 16w+16).
// =====================================================================
__global__ __launch_bounds__(256) void k_mlp_wmma(
    const float* __restrict__ pred, const float* __restrict__ W1,
    const float* __restrict__ b1, const float* __restrict__ W2,
    const float* __restrict__ b2, float* __restrict__ out)
{
  __shared__ float h[WAVES][16][32];
  const int tid = threadIdx.x, lane = tid & 31, wave = tid >> 5;
  const int m0 = wave * 16;
  const bool lo = lane < 16;
  const int l16 = lane & 15;

  const v8f czero = {0.f,0.f,0.f,0.f,0.f,0.f,0.f,0.f};

  // A (16x4): M=lane%16, K=vgpr + 2*(lane>=16); only K=0 column non-zero
  v2f a; a.x = lo ? pred[m0 + l16] : 0.f; a.y = 0.f;

  // GEMM1: h[m, n] = pred[m] * W1[n];  two WMMAs cover hidden 0..15 / 16..31
  #pragma unroll
  for (int half = 0; half < 2; ++half) {
    const int noff = half * 16;
    v2f bb; bb.x = lo ? W1[noff + l16] : 0.f; bb.y = 0.f;   // B[0][n] = W1[n]
    v8f c = __builtin_amdgcn_wmma_f32_16x16x4_f32(
        false, a, false, bb, (short)0, czero, false, false);
    const float bias = b1[noff + l16];
    #pragma unroll
    for (int j = 0; j < 8; ++j) {
      float hv = c[j] + bias;               // D: M = j + 8*(lane>=16), N = lane%16
      hv = hv > 0.f ? hv : 0.f;
      h[wave][j + (lo ? 0 : 8)][noff + l16] = hv;
    }
  }
  __syncthreads();   // transpose h through LDS into A-operand layout

  // GEMM2: out[m] = sum_k h[m,k] * W2[k]; K=32 as 8 accumulating WMMAs.
  // B columns replicated with W2 so every N column holds the result; read N=0.
  v8f acc = czero;
  #pragma unroll
  for (int i = 0; i < 8; ++i) {
    const int k0 = 4 * i;
    v2f a2, bv;
    const int ka = k0 + (lo ? 0 : 2);
    a2.x = h[wave][l16][ka + 0];
    a2.y = h[wave][l16][ka + 1];
    bv.x = W2[ka + 0];
    bv.y = W2[ka + 1];
    acc = __builtin_amdgcn_wmma_f32_16x16x4_f32(
        false, a2, false, bv, (short)0, acc, false, false);
  }

  if (l16 == 0) {                            // lanes 0 / 16 hold column N=0
    const float bias2 = b2[0];
    #pragma unroll
    for (int j = 0; j < 8; ++j)
      out[m0 + j + (lo ? 0 : 8)] = acc[j] + bias2;
  }
}

// =====================================================================
extern "C" void kernel_launch(void* const* d_in, const int* in_sizes, int n_in,
                              void* d_out, int out_size, void* d_ws, size_t ws_size,
                              hipStream_t stream)
{
  const float* q    = (const float*)d_in[0];
  const float* X    = (const float*)d_in[1];
  const float* y    = (const float*)d_in[2];
  const float* fw   = (const float*)d_in[3];
  const float* beta = (const float*)d_in[4];
  const float* W1   = (const float*)d_in[5];
  const float* b1   = (const float*)d_in[6];
  const float* W2   = (const float*)d_in[7];
  const float* b2   = (const float*)d_in[8];
  float* out = (float*)d_out;

  char* ws = (char*)d_ws;
  float* candVal = (float*)ws;                                   // 4 MB
  int*   candIdx = (int*)(ws + (size_t)B_ * NCAND * 4);          // 4 MB
  float* pred    = (float*)(ws + (size_t)B_ * NCAND * 8);        // 512 B

  k_dist_topk<<<dim3(CHUNKS, 4), 256, 0, stream>>>(q, X, fw, candVal, candIdx);
  k_select   <<<dim3(B_),       256, 0, stream>>>(candVal, candIdx, y, beta, pred);
  k_mlp_wmma <<<dim3(1),        256, 0, stream>>>(pred, W1, b1, W2, b2, out);
}